// VectorQuantizer_69724499083489
// MI455X (gfx1250) — compile-verified
//
#include <hip/hip_runtime.h>

typedef float v2f __attribute__((ext_vector_type(2)));
typedef float v8f __attribute__((ext_vector_type(8)));

#define D        64
#define KCODES   1024
#define NROWS    65536
#define CHUNK    128            // codes staged in LDS at a time
#define SWS      68             // padded LDS stride (floats): 64 data + 4 pad -> conflict-free B reads
#define ROWS_PB  128            // 8 waves * 16 rows
#define NTILES   (CHUNK / 16)   // 8 column tiles per chunk (even, required by 2x pipeline)
#define NCHUNKS  (KCODES / CHUNK)

__global__ __launch_bounds__(256)
void vq_main_kernel(const float* __restrict__ x,
                    const float* __restrict__ w,
                    float* __restrict__ outQ,
                    float* __restrict__ outLoss,
                    float* __restrict__ outIdx)
{
    __shared__ float sw[CHUNK * SWS];   // staged codebook chunk (bank-padded)
    __shared__ float shn[KCODES];       // 0.5*||w_k||^2
    __shared__ float sred[256];

    const int tid  = threadIdx.x;
    const int wave = tid >> 5;
    const int lane = tid & 31;
    const int lh   = lane >> 4;     // half-wave selects K-pair / row-group per ISA layout
    const int ll   = lane & 15;
    const int koff = 2 * lh;
    const int rowBase = blockIdx.x * ROWS_PB + wave * 16;
    const int myRow   = rowBase + ll;

    // --- precompute half-norms of all codes (codebook is L2-hot) ---
    for (int k = tid; k < KCODES; k += 256) {
        const float4* wr = (const float4*)(w + k * D);
        float s = 0.f;
        #pragma unroll
        for (int i = 0; i < D / 4; ++i) {
            float4 v = wr[i];
            s += v.x * v.x + v.y * v.y + v.z * v.z + v.w * v.w;
        }
        shn[k] = 0.5f * s;
    }

    // --- preload A fragments: 16 rows x D=64 as 16 chunks of 16x4 f32 ---
    // ISA 32-bit A 16x4 layout: lane<16 holds (M=ll, K=c*4+0,1); lane>=16 holds (M=ll, K=c*4+2,3)
    v2f afrag[16];
    #pragma unroll
    for (int c = 0; c < 16; ++c)
        afrag[c] = *(const v2f*)(x + myRow * D + c * 4 + koff);

    float best[8];
    int   bidx[8];
    #pragma unroll
    for (int r = 0; r < 8; ++r) { best[r] = -3.4e38f; bidx[r] = 0; }

    // B-fragment loader: ISA 32-bit B 4x16 layout: lane holds (N=ll, K=c*4+koff, +1)
    auto loadB = [&](v2f* dst, int t) {
        const float* bb = sw + (t * 16 + ll) * SWS + koff;
        #pragma unroll
        for (int c = 0; c < 16; ++c)
            dst[c] = *(const v2f*)(bb + c * 4);
    };

    for (int ch = 0; ch < NCHUNKS; ++ch) {
        __syncthreads();   // previous chunk's LDS reads done (covers shn on ch==0)
        // stage CHUNK codes -> LDS with padded stride (float4 granules, 16B-aligned)
        for (int i = tid; i < CHUNK * 16; i += 256) {
            int code = i >> 4, seg = i & 15;
            *(float4*)(sw + code * SWS + seg * 4) =
                *(const float4*)(w + (ch * CHUNK + code) * D + seg * 4);
        }
        __syncthreads();

        // software-pipelined tile loop: B loads for the next tile are issued
        // before the current tile's WMMA chain, so LDS latency hides under WMMA
        v2f bA[16], bB[16];
        loadB(bA, 0);
        for (int t = 0; t < NTILES; t += 2) {
            loadB(bB, t + 1);                          // prefetch odd tile
            {
                v8f acc = {};
                #pragma unroll
                for (int c = 0; c < 16; ++c)
                    acc = __builtin_amdgcn_wmma_f32_16x16x4_f32(
                              false, afrag[c], false, bA[c], (short)0, acc, false, false);
                const int col = ch * CHUNK + t * 16 + ll;
                const float hn = shn[col];
                #pragma unroll
                for (int r = 0; r < 8; ++r) {
                    float s = acc[r] - hn;             // x.w - 0.5||w||^2 ; argmax == argmin dist
                    if (s > best[r]) { best[r] = s; bidx[r] = col; }
                }
            }
            if (t + 2 < NTILES) loadB(bA, t + 2);      // prefetch next even tile
            {
                v8f acc = {};
                #pragma unroll
                for (int c = 0; c < 16; ++c)
                    acc = __builtin_amdgcn_wmma_f32_16x16x4_f32(
                              false, afrag[c], false, bB[c], (short)0, acc, false, false);
                const int col = ch * CHUNK + (t + 1) * 16 + ll;
                const float hn = shn[col];
                #pragma unroll
                for (int r = 0; r < 8; ++r) {
                    float s = acc[r] - hn;
                    if (s > best[r]) { best[r] = s; bidx[r] = col; }
                }
            }
        }
    }

    // --- cross-lane argmax over the 16 lanes sharing each row; then outputs ---
    float sumsq = 0.f;
    #pragma unroll
    for (int r = 0; r < 8; ++r) {
        float s = best[r];
        int   bi = bidx[r];
        #pragma unroll
        for (int m = 1; m < 16; m <<= 1) {
            float so = __shfl_xor(s, m, 32);
            int   io = __shfl_xor(bi, m, 32);
            if (so > s || (so == s && io < bi)) { s = so; bi = io; }  // first-max tie-break
        }
        const int row = rowBase + lh * 8 + r;
        if (ll == 0) outIdx[row] = (float)bi;

        // gather winning code, emit straight-through output (4 consecutive floats/lane)
        float4 q  = *(const float4*)(w + bi * D + ll * 4);
        float4 xv = *(const float4*)(x + row * D + ll * 4);
        float dx = q.x - xv.x, dy = q.y - xv.y, dz = q.z - xv.z, dw = q.w - xv.w;
        float4 st;
        st.x = xv.x + dx; st.y = xv.y + dy; st.z = xv.z + dz; st.w = xv.w + dw;
        *(float4*)(outQ + row * D + ll * 4) = st;
        sumsq += dx * dx + dy * dy + dz * dz + dw * dw;
    }

    // --- block reduction of squared error -> loss (pre-scaled by BETA / (N*D)) ---
    sred[tid] = sumsq;
    __syncthreads();
    for (int s = 128; s > 0; s >>= 1) {
        if (tid < s) sred[tid] += sred[tid + s];
        __syncthreads();
    }
    if (tid == 0)
        atomicAdd(outLoss, sred[0] * (0.25f / 4194304.0f));
}

__global__ void vq_init_kernel(float* __restrict__ loss)
{
    if (threadIdx.x == 0 && blockIdx.x == 0) *loss = 0.f;
}

__global__ void vq_wt_kernel(const float* __restrict__ w, float* __restrict__ wt)
{
    int i = blockIdx.x * 256 + threadIdx.x;   // 65536 elements of w^T [64,1024]
    int d = i >> 10;
    int k = i & 1023;
    wt[i] = w[k * D + d];
}

extern "C" void kernel_launch(void* const* d_in, const int* in_sizes, int n_in,
                              void* d_out, int out_size, void* d_ws, size_t ws_size,
                              hipStream_t stream)
{
    (void)in_sizes; (void)n_in; (void)out_size; (void)d_ws; (void)ws_size;
    const float* x = (const float*)d_in[0];   // [64,32,32,64] -> [65536,64]
    const float* w = (const float*)d_in[1];   // [1024,64]

    float* out     = (float*)d_out;
    float* outQ    = out;                          // 4194304 floats
    float* outLoss = out + 4194304;                // 1 float
    float* outIdx  = out + 4194305;                // 65536 (indices as float)
    float* outWT   = out + 4194305 + 65536;        // 65536 (w^T, [64,1024])

    vq_init_kernel<<<1, 1, 0, stream>>>(outLoss);
    vq_wt_kernel<<<256, 256, 0, stream>>>(w, outWT);
    vq_main_kernel<<<NROWS / ROWS_PB, 256, 0, stream>>>(x, w, outQ, outLoss, outIdx);
}